// SoftMorphology_16329465659782
// MI455X (gfx1250) — compile-verified
//
#include <hip/hip_runtime.h>
#include <stdint.h>

// Soft-skeleton (soft_erode / soft_open thinning), fused one-kernel-per-round.
//   e_{k+1} = erode(e_k)  (5-pt cross min)
//   delta_k = relu(e_k - dilate3x3(e_{k+1}))
//   k==0 : skel = delta ;  k>=1 : skel += relu(delta - skel*delta)
// Memory-bound stencil (far below 1 flop/byte): WMMA inapplicable (min/max),
// so the CDNA5-specific path used is async global->LDS staging (ASYNCcnt,
// b128 granularity) + LDS halo reuse. ~256 MiB traffic x 11 rounds, working
// set ~256 MB vs 192 MB L2 -> effectively L2-bandwidth bound.

#define IMG_H 1024
#define IMG_W 1024
#define TX 64
#define TY 32
#define NTHREADS 256
#define SINW 72        // staged input width (floats): gx0-4 .. gx0+67, 16B aligned
#define SINH 36        // staged input rows: gy0-2 .. gy0+33
#define ERW (TX + 2)   // 66  (eroded tile width,  halo 1)
#define ERH (TY + 2)   // 34  (eroded tile height, halo 1)

// Flat (generic) shared pointer -> LDS byte offset: low 32 bits of the LDS
// aperture address ARE the wave-relative LDS offset (ISA 10.2 aperture calc).
__device__ __forceinline__ uint32_t lds_off_of(const void* p) {
    return (uint32_t)(uintptr_t)p;
}

__device__ __forceinline__ int clampi(int v, int lo, int hi) {
    return v < lo ? lo : (v > hi ? hi : v);
}

__global__ __launch_bounds__(NTHREADS) void
softskel_step_kernel(const float* __restrict__ ein,
                     float* __restrict__ eout,
                     float* __restrict__ skel,
                     int first)
{
    __shared__ float s_in[SINH * SINW];  // e_k tile + halo
    __shared__ float s_er[ERH * ERW];    // e_{k+1} tile + halo 1

    const int tid = threadIdx.x;
    const int gx0 = blockIdx.x * TX;
    const int gy0 = blockIdx.y * TY;
    const size_t base = (size_t)blockIdx.z * (size_t)(IMG_H * IMG_W);
    const float* einb = ein + base;

    // ---- Phase 1: async global->LDS staging of clamped halo tile ----------
    // y-clamping preserves row contiguity, so interior-x tiles use b128 async
    // copies (4x fewer vmem issues). x-border tiles use clamped b32 gathers.
    // (clamp-to-edge == identity padding for a single stride-1 min/max pool)
    if (blockIdx.x != 0 && blockIdx.x != gridDim.x - 1) {
        // 36 rows x 18 float4 = 648 vector elements
        for (int idx = tid; idx < SINH * (SINW / 4); idx += NTHREADS) {
            int r  = idx / (SINW / 4);
            int c4 = idx - r * (SINW / 4);
            int gy = clampi(gy0 - 2 + r, 0, IMG_H - 1);
            unsigned long long gp =
                (unsigned long long)(einb + (size_t)gy * IMG_W + (gx0 - 4 + c4 * 4));
            uint32_t lo = lds_off_of(&s_in[r * SINW + c4 * 4]);
            asm volatile("global_load_async_to_lds_b128 %0, %1, off"
                         :: "v"(lo), "v"(gp) : "memory");
        }
    } else {
        for (int idx = tid; idx < SINH * SINW; idx += NTHREADS) {
            int r = idx / SINW;
            int c = idx - r * SINW;
            int gy = clampi(gy0 - 2 + r, 0, IMG_H - 1);
            int gx = clampi(gx0 - 4 + c, 0, IMG_W - 1);
            unsigned long long gp =
                (unsigned long long)(einb + (size_t)gy * IMG_W + gx);
            uint32_t lo = lds_off_of(&s_in[idx]);
            asm volatile("global_load_async_to_lds_b32 %0, %1, off"
                         :: "v"(lo), "v"(gp) : "memory");
        }
    }
    asm volatile("s_wait_asynccnt 0" ::: "memory");
    __syncthreads();   // all waves' async LDS writes visible

    // ---- Phase 2: erode (5-pt cross min) into ERH x ERW -------------------
    // s_er(r,c) is e_{k+1} at (gy0-1+r, gx0-1+c); center in s_in is (r+1,c+3).
    // Out-of-image positions must be -inf (identity for the following max):
    // clamping is NOT valid across composed pooling stages.
    for (int idx = tid; idx < ERH * ERW; idx += NTHREADS) {
        int r = idx / ERW;
        int c = idx - r * ERW;
        int gy = gy0 - 1 + r;
        int gx = gx0 - 1 + c;
        float v;
        if (gy < 0 || gy >= IMG_H || gx < 0 || gx >= IMG_W) {
            v = -INFINITY;
        } else {
            const float* p = &s_in[(r + 1) * SINW + (c + 3)];
            float m0 = fminf(p[0],     p[-SINW]);
            float m1 = fminf(p[SINW],  p[-1]);
            v = fminf(fminf(m0, m1), p[1]);
        }
        s_er[idx] = v;
    }
    __syncthreads();

    // ---- Phase 3: dilate 3x3 + skel update, 4 contiguous px/thread/pass ---
    const int ty_base = tid >> 4;          // 0..15
    const int tx      = (tid & 15) << 2;   // 0,4,...,60
    #pragma unroll
    for (int half = 0; half < 2; ++half) {
        int ty = ty_base + half * 16;
        const float* e0 = &s_er[ty * ERW + tx];   // top-left of 3x6 region

        float a0=e0[0],       a1=e0[1],       a2=e0[2],       a3=e0[3],       a4=e0[4],       a5=e0[5];
        float b0=e0[ERW],     b1=e0[ERW+1],   b2=e0[ERW+2],   b3=e0[ERW+3],   b4=e0[ERW+4],   b5=e0[ERW+5];
        float c0=e0[2*ERW],   c1=e0[2*ERW+1], c2=e0[2*ERW+2], c3=e0[2*ERW+3], c4=e0[2*ERW+4], c5=e0[2*ERW+5];

        // vertical max of 3, then horizontal max of 3 (shared across 4 px)
        float v0 = fmaxf(fmaxf(a0, b0), c0);
        float v1 = fmaxf(fmaxf(a1, b1), c1);
        float v2 = fmaxf(fmaxf(a2, b2), c2);
        float v3 = fmaxf(fmaxf(a3, b3), c3);
        float v4 = fmaxf(fmaxf(a4, b4), c4);
        float v5 = fmaxf(fmaxf(a5, b5), c5);
        float m0 = fmaxf(fmaxf(v0, v1), v2);
        float m1 = fmaxf(fmaxf(v1, v2), v3);
        float m2 = fmaxf(fmaxf(v2, v3), v4);
        float m3 = fmaxf(fmaxf(v3, v4), v5);

        // x = e_k centers (aligned b128 LDS load)
        float4 xv = *(const float4*)&s_in[(ty + 2) * SINW + (tx + 4)];
        float4 d;
        d.x = fmaxf(xv.x - m0, 0.0f);
        d.y = fmaxf(xv.y - m1, 0.0f);
        d.z = fmaxf(xv.z - m2, 0.0f);
        d.w = fmaxf(xv.w - m3, 0.0f);

        size_t gidx = base + (size_t)(gy0 + ty) * IMG_W + (gx0 + tx);

        // e_{k+1} centers are the middle-row values already in registers
        *(float4*)&eout[gidx] = make_float4(b1, b2, b3, b4);

        if (first) {
            *(float4*)&skel[gidx] = d;
        } else {
            float4 s = *(const float4*)&skel[gidx];
            s.x = s.x + fmaxf(d.x - s.x * d.x, 0.0f);
            s.y = s.y + fmaxf(d.y - s.y * d.y, 0.0f);
            s.z = s.z + fmaxf(d.z - s.z * d.z, 0.0f);
            s.w = s.w + fmaxf(d.w - s.w * d.w, 0.0f);
            *(float4*)&skel[gidx] = s;
        }
    }
}

extern "C" void kernel_launch(void* const* d_in, const int* in_sizes, int n_in,
                              void* d_out, int out_size, void* d_ws, size_t ws_size,
                              hipStream_t stream) {
    const float* img  = (const float*)d_in[0];
    float*       skel = (float*)d_out;

    const int batch = in_sizes[0] / (IMG_H * IMG_W);   // 16
    const size_t N  = (size_t)batch * IMG_H * IMG_W;

    // Ping-pong eroded-image buffers in workspace (needs 2*N*4 = 128 MiB).
    float* eA = (float*)d_ws;
    float* eB = eA + N;

    dim3 grid(IMG_W / TX, IMG_H / TY, batch);
    dim3 block(NTHREADS);

    const int num_iter = 10;   // static in the reference (graph-capture safe)
    const float* ein = img;
    for (int k = 0; k <= num_iter; ++k) {   // k==0 is the skel init round
        float* eout = (k & 1) ? eB : eA;
        softskel_step_kernel<<<grid, block, 0, stream>>>(ein, eout, skel, k == 0);
        ein = eout;
    }
}